// TritonLinearHead_79121887527346
// MI455X (gfx1250) — compile-verified
//
#include <hip/hip_runtime.h>

typedef __attribute__((ext_vector_type(2))) float v2f;
typedef __attribute__((ext_vector_type(4))) float v4f;
typedef __attribute__((ext_vector_type(8))) float v8f;

#define C_DIM 64
#define P_DIM 1024
#define Q_DIM 1024

__global__ __launch_bounds__(256) void linear_head_wmma_kernel(
    const float* __restrict__ x, const float* __restrict__ w, float* __restrict__ y)
{
  // One block per (n, p) row.
  const int np = blockIdx.x;
  const int n  = np >> 10;       // / P_DIM
  const int p  = np & (P_DIM - 1);

  __shared__ float wlds[C_DIM * C_DIM];  // 16 KB, row-major [d][c]

  const int tid = threadIdx.x;

  // Cooperative staging of W: 4096 floats via b128. Regular-temporal: W is
  // reused by every block, keep it resident in L2.
  {
    const v4f* wsrc = (const v4f*)w;
    v4f*       wdst = (v4f*)wlds;
#pragma unroll
    for (int i = 0; i < 4; ++i)
      wdst[tid + 256 * i] = wsrc[tid + 256 * i];
  }
  __syncthreads();

  const int wave   = tid >> 5;   // 0..7
  const int lane   = tid & 31;
  const int lane16 = lane & 15;
  const int laneHi = lane >> 4;  // 0 or 1 (half-wave select)

  const size_t plane = (size_t)P_DIM * Q_DIM;  // elements per (n,c) plane
  const float* xrow = x + ((size_t)n * C_DIM) * plane + (size_t)p * Q_DIM;
  float*       yrow = y + ((size_t)n * C_DIM) * plane + (size_t)p * Q_DIM;

  // Causal: only q-tiles at-or-below the diagonal need compute.
  const int ntiles = (p >> 4) + 1;

  for (int t = wave; t < ntiles; t += 8) {   // wave-uniform loop: EXEC all-1s
    const int qb = t << 4;
    const int q  = qb + lane16;

    // B fragments: 4x16 (K x N) slice of x, K striped per half-wave.
    // lane holds B[2*laneHi + r][lane16] in vgpr r  ->  x[c = k0+2*laneHi+r][q]
    // x is single-use -> non-temporal loads (TH=NT), don't pollute L2.
    v2f bfrag[16];
#pragma unroll
    for (int ks = 0; ks < 16; ++ks) {
      const int c0 = (ks << 2) + 2 * laneHi;
      v2f b;
      b.x = __builtin_nontemporal_load(&xrow[(size_t)c0       * plane + q]);
      b.y = __builtin_nontemporal_load(&xrow[(size_t)(c0 + 1) * plane + q]);
      bfrag[ks] = b;
    }

    v8f acc[4] = {};  // 4 d-tiles of 16x16 f32 accumulators

#pragma unroll
    for (int ks = 0; ks < 16; ++ks) {
      const int c0 = (ks << 2) + 2 * laneHi;
#pragma unroll
      for (int dt = 0; dt < 4; ++dt) {
        // A fragment: 16x4 tile of W. lanes 0-15: M=lane16, K={c0,c0+1};
        // lanes 16-31: M=lane16, K={c0+2,c0+3}. b64 read from LDS.
        const v2f a = *(const v2f*)&wlds[(dt * 16 + lane16) * C_DIM + c0];
        acc[dt] = __builtin_amdgcn_wmma_f32_16x16x4_f32(
            /*neg_a=*/false, a, /*neg_b=*/false, bfrag[ks],
            /*c_mod=*/(short)0, acc[dt], /*reuse_a=*/false, /*reuse_b=*/false);
      }
    }

    // Store: C/D layout => vgpr v, lanes 0-15: M=v; lanes 16-31: M=v+8.
    // d = dt*16 + v + 8*laneHi, q = qb + lane16. Predicate diagonal tile.
    // y is write-once -> non-temporal stores.
    if (q <= p) {
#pragma unroll
      for (int dt = 0; dt < 4; ++dt) {
#pragma unroll
        for (int v = 0; v < 8; ++v) {
          const int d = dt * 16 + v + 8 * laneHi;
          __builtin_nontemporal_store(acc[dt][v], &yrow[(size_t)d * plane + q]);
        }
      }
    }
  }

  // Zero-fill the masked region q in [p+1, Q) for all 64 output channels.
  // Vectorized: scalar head to 16B alignment, then b128 non-temporal stores.
  const int q0 = p + 1;
  if (q0 < Q_DIM) {
    const int qa   = (q0 + 3) & ~3;        // 16B-aligned start (<= Q_DIM)
    const int head = qa - q0;              // 0..3 scalar elements
    const int nvec = (Q_DIM - qa) >> 2;    // number of b128 stores per row
    const v4f z4 = {0.f, 0.f, 0.f, 0.f};
    for (int d = 0; d < C_DIM; ++d) {
      float* row = yrow + (size_t)d * plane;
      if (tid < head)
        __builtin_nontemporal_store(0.0f, &row[q0 + tid]);
      for (int i = tid; i < nvec; i += 256)
        __builtin_nontemporal_store(z4, (v4f*)(row + qa + 4 * i));
    }
  }
}

extern "C" void kernel_launch(void* const* d_in, const int* in_sizes, int n_in,
                              void* d_out, int out_size, void* d_ws, size_t ws_size,
                              hipStream_t stream) {
  const float* x = (const float*)d_in[0];
  const float* w = (const float*)d_in[1];
  float*       y = (float*)d_out;

  const int n_batch = in_sizes[0] / (C_DIM * P_DIM * Q_DIM);  // N = 2
  const int grid    = n_batch * P_DIM;                        // one block per (n,p)

  linear_head_wmma_kernel<<<grid, 256, 0, stream>>>(x, w, y);
}